// EdgeFeature_23794118819863
// MI455X (gfx1250) — compile-verified
//
#include <hip/hip_runtime.h>

// Problem constants (from reference): point_cloud (4, 64, 4096) f32, K=16
#define B_  4
#define D_  64
#define N_  4096
#define K_  16
#define CT_ (N_ / 16)     // 256 column tiles of 16 points

typedef __attribute__((ext_vector_type(2))) float v2f;
typedef __attribute__((ext_vector_type(8))) float v8f;

// Low 32 bits of a generic LDS pointer == LDS byte offset (aperture mapping
// truncates to addr[31:0]); that is what async-to-LDS VDST expects.
__device__ __forceinline__ uint32_t lds_off(const void* p) {
    return (uint32_t)(uintptr_t)p;
}

// ---------------------------------------------------------------------------
// Kernel 1: squared norms  sq[b][n] = sum_d x[b][d][n]^2   (float4 vectorized)
// ---------------------------------------------------------------------------
__global__ void __launch_bounds__(256)
sq_kernel(const float* __restrict__ x, float* __restrict__ sq) {
    int id = blockIdx.x * blockDim.x + threadIdx.x;   // B_*N_/4 threads
    int b  = id / (N_ / 4);
    int n4 = (id - b * (N_ / 4)) * 4;
    const float* xp = x + (size_t)b * D_ * N_ + n4;
    float4 s = make_float4(0.f, 0.f, 0.f, 0.f);
#pragma unroll
    for (int d = 0; d < D_; ++d) {
        float4 v = *(const float4*)(xp + (size_t)d * N_);
        s.x = fmaf(v.x, v.x, s.x);
        s.y = fmaf(v.y, v.y, s.y);
        s.z = fmaf(v.z, v.z, s.z);
        s.w = fmaf(v.w, v.w, s.w);
    }
    *(float4*)(sq + (size_t)b * N_ + n4) = s;
}

// ---------------------------------------------------------------------------
// Async stage of one 64x16 f32 column tile into an LDS buffer.
// 128 threads x 2 B128 transfers = 4 KB, direct memory->LDS (ASYNCcnt).
// ---------------------------------------------------------------------------
__device__ __forceinline__ void stage_tile_async(const float* __restrict__ xb,
                                                 int colbase,
                                                 float (*BsBuf)[16], int tid) {
#pragma unroll
    for (int s = 0; s < 2; ++s) {
        int q  = tid + s * 128;        // quad id 0..255
        int d  = q >> 2;               // dim 0..63
        int c4 = (q & 3) * 4;          // col quad 0,4,8,12
        uint32_t lo = lds_off(&BsBuf[d][c4]);
        uint32_t vo = (uint32_t)(((d * N_) + colbase + c4) * sizeof(float));
        asm volatile("global_load_async_to_lds_b128 %0, %1, %2"
                     :: "v"(lo), "v"(vo), "s"(xb)
                     : "memory");
    }
}

// ---------------------------------------------------------------------------
// Kernel 2: fused f32-WMMA Gram matrix + top-K selection.
// 128-thread block (4 waves) owns 64 query rows; wave w owns 16 rows.
// Double-buffered async LDS staging overlaps the next column tile with the
// current tile's WMMA chain + insertion. One block barrier per tile.
// ---------------------------------------------------------------------------
__global__ void __launch_bounds__(128)
knn_kernel(const float* __restrict__ x, const float* __restrict__ sq,
           int* __restrict__ idxws, float* __restrict__ out_idx) {
    __shared__ float Bs[2][D_][16];       // double-buffered column tiles
    __shared__ float sqBw[4][16];         // per-wave column norms
    __shared__ float gT[4][16][16];       // per-wave Gram tile
    __shared__ float bestD[64][K_];       // sorted ascending distance lists
    __shared__ int   bestI[64][K_];

    const int tid    = threadIdx.x;
    const int wave   = tid >> 5;
    const int lane   = tid & 31;
    const int khalf  = lane >> 4;         // K-half of the WMMA operand
    const int lanelo = lane & 15;

    const int wg      = blockIdx.x;               // B_ * (N_/64) blocks
    const int b       = wg / (N_ / 64);
    const int rowblk  = (wg - b * (N_ / 64)) * 64;
    const int rowbase = rowblk + wave * 16;

    const float* xb = x + (size_t)b * D_ * N_;

    // Preload A fragments for all 16 K-steps (dims 0..63), rows rowbase..+15.
    // A(16x4) f32 layout: lane m=L&15; vgpr j holds K = 4*kk + 2*(L>>4) + j.
    v2f aF[16];
#pragma unroll
    for (int kk = 0; kk < 16; ++kk) {
        int d0 = kk * 4 + 2 * khalf;
        aF[kk].x = xb[(size_t)d0       * N_ + rowbase + lanelo];
        aF[kk].y = xb[(size_t)(d0 + 1) * N_ + rowbase + lanelo];
    }

    // Init top-K lists (FLT_MAX sentinels).
    if (lane < 16) {
        int r = wave * 16 + lane;
#pragma unroll
        for (int k = 0; k < K_; ++k) { bestD[r][k] = 3.4e38f; bestI[r][k] = 0; }
    }
    const float srow = sq[b * N_ + rowbase + lanelo];   // |x_row|^2

    // Prologue: stage tile 0 into buffer 0.
    stage_tile_async(xb, 0, Bs[0], tid);

    int p = 0;
    for (int ct = 0; ct < CT_; ++ct) {
        const int colbase = ct * 16;

        // Our async stores (issued last iteration, or prologue) have landed;
        // barrier makes every wave's stores visible to everyone.
        asm volatile("s_wait_asynccnt 0x0" ::: "memory");
        __syncthreads();

        // Kick async stage of the next tile into the other buffer; it runs
        // concurrently with this tile's WMMA chain and insertion phase.
        stage_tile_async(xb, ((ct + 1) & (CT_ - 1)) * 16, Bs[p ^ 1], tid);

        // Per-wave column norms for this tile (wave-local consumption).
        if (lane < 16)
            sqBw[wave][lane] = sq[b * N_ + colbase + lane];

        // Gram tile: acc = sum_k A(16x4) * B(4x16), full f32 WMMA chain.
        v8f acc = {};
#pragma unroll
        for (int kk = 0; kk < 16; ++kk) {
            int d0 = kk * 4 + 2 * khalf;
            v2f bF;
            bF.x = Bs[p][d0][lanelo];
            bF.y = Bs[p][d0 + 1][lanelo];
            acc = __builtin_amdgcn_wmma_f32_16x16x4_f32(
                      false, aF[kk], false, bF, (short)0, acc, false, false);
        }

        // Spill tile to LDS: vgpr i -> row (i + 8*khalf), col lanelo.
#pragma unroll
        for (int i = 0; i < 8; ++i)
            gT[wave][i + 8 * khalf][lanelo] = acc[i];

        // Wave-local visibility only (producer == consumer wave): a DS-counter
        // drain suffices in lockstep wave32; no block barrier needed.
        asm volatile("s_wait_dscnt 0x0" ::: "memory");

        // Stable insertion: lanes 0..15 each own one row of this wave.
        if (lane < 16) {
            const int r    = lane;
            const int grow = rowbase + r;
            float* bd = bestD[wave * 16 + r];
            int*   bi = bestI[wave * 16 + r];
            for (int c = 0; c < 16; ++c) {
                const int gn = colbase + c;
                if (gn == grow) continue;              // exclude self (rank-0)
                float g  = gT[wave][r][c];
                float d2 = srow + sqBw[wave][c] - 2.0f * g;
                float dd = sqrtf(fmaxf(d2, 0.0f));
                if (dd < bd[K_ - 1]) {                 // strict: keep earliest on ties
                    int j = K_ - 1;
                    while (j > 0 && bd[j - 1] > dd) {  // strict: stable order
                        bd[j] = bd[j - 1]; bi[j] = bi[j - 1]; --j;
                    }
                    bd[j] = dd; bi[j] = gn;
                }
            }
        }
        p ^= 1;
    }
    __syncthreads();

    // Emit neighbor indices: int workspace for the gather kernel, and the
    // second tuple output (as float) directly.
    if (lane < 16) {
        const int grow = rowbase + lane;
        const int r    = wave * 16 + lane;
#pragma unroll
        for (int k = 0; k < K_; ++k) {
            int v = bestI[r][k];
            idxws[(b * N_ + grow) * K_ + k]   = v;
            out_idx[(b * N_ + grow) * K_ + k] = (float)v;
        }
    }
}

// ---------------------------------------------------------------------------
// Kernel 3: edge features, fully coalesced 134 MB output stream.
// Thread (b,c,n) writes 16 consecutive floats as 4x float4.
//   c <  64: broadcast central point
//   c >= 64: neighbor gather minus central
// ---------------------------------------------------------------------------
__global__ void __launch_bounds__(256)
edge_kernel(const float* __restrict__ x, const int* __restrict__ idxws,
            float* __restrict__ out) {
    int id = blockIdx.x * blockDim.x + threadIdx.x;  // B_*2*D_*N_ threads
    int n  = id % N_;
    int c  = (id / N_) % (2 * D_);
    int b  = id / (N_ * 2 * D_);
    int cc = c & (D_ - 1);

    const float* xr = x + (size_t)(b * D_ + cc) * N_;
    const float central = xr[n];

    float4 v[4];
    if (c < D_) {
        float4 f = make_float4(central, central, central, central);
        v[0] = f; v[1] = f; v[2] = f; v[3] = f;
    } else {
        const int* ip = idxws + (size_t)(b * N_ + n) * K_;
#pragma unroll
        for (int q = 0; q < 4; ++q) {
            float4 f;
            f.x = xr[ip[q * 4 + 0]] - central;
            f.y = xr[ip[q * 4 + 1]] - central;
            f.z = xr[ip[q * 4 + 2]] - central;
            f.w = xr[ip[q * 4 + 3]] - central;
            v[q] = f;
        }
    }
    float4* op = (float4*)(out + (size_t)id * K_);
    op[0] = v[0]; op[1] = v[1]; op[2] = v[2]; op[3] = v[3];
}

// ---------------------------------------------------------------------------
extern "C" void kernel_launch(void* const* d_in, const int* in_sizes, int n_in,
                              void* d_out, int out_size, void* d_ws, size_t ws_size,
                              hipStream_t stream) {
    const float* x = (const float*)d_in[0];
    float* out = (float*)d_out;

    // Workspace layout: [ sq: B*N floats ][ idx: B*N*K ints ]
    float* sq    = (float*)d_ws;
    int*   idxws = (int*)((char*)d_ws + (size_t)B_ * N_ * sizeof(float));

    // Output layout: [ edge_feature: B*2D*N*K ][ idx (as float): B*N*K ]
    float* out_idx = out + (size_t)B_ * 2 * D_ * N_ * K_;

    sq_kernel  <<<(B_ * N_ / 4) / 256,      256, 0, stream>>>(x, sq);
    knn_kernel <<<B_ * (N_ / 64),           128, 0, stream>>>(x, sq, idxws, out_idx);
    edge_kernel<<<(B_ * 2 * D_ * N_) / 256, 256, 0, stream>>>(x, idxws, out);
}